// MeshPartEncoder_68410239090854
// MI455X (gfx1250) — compile-verified
//
#include <hip/hip_runtime.h>
#include <math.h>

typedef __attribute__((ext_vector_type(2))) float v2f;
typedef __attribute__((ext_vector_type(8))) float v8f;

#define NNODES 20000
#define NEDGES 320000
#define ETOT   (NEDGES + NNODES)
#define CH     128

// ---------- helpers ----------
__device__ __forceinline__ unsigned fflip(float f) {
  unsigned u = __float_as_uint(f);
  return (u & 0x80000000u) ? ~u : (u | 0x80000000u);
}
__device__ __forceinline__ float funflip(unsigned u) {
  return __uint_as_float((u & 0x80000000u) ? (u & 0x7FFFFFFFu) : ~u);
}
__device__ __forceinline__ void edge_st(const int* __restrict__ ei, int e, int& s, int& t) {
  if (e < NEDGES) { s = ei[e]; t = ei[NEDGES + e]; }
  else            { s = e - NEDGES; t = s; }
}

__global__ void fill_f32(float* p, float v, int n) {
  int i = blockIdx.x * blockDim.x + threadIdx.x;
  if (i < n) p[i] = v;
}
__global__ void fill_u32(unsigned* p, unsigned v, int n) {
  int i = blockIdx.x * blockDim.x + threadIdx.x;
  if (i < n) p[i] = v;
}

// ---------- fp32 WMMA GEMM: C[M,Ncol] = A[M,K] @ W[K,Ncol] + bias ----------
// block = 128 threads = 4 waves; each wave owns a 16x16 tile, block covers 64 rows x 16 cols.
// B column panel (K x 16) staged in LDS once per block. K is compile-time so the
// k-loop fully unrolls; for K%4==0 all loads are unguarded (no exec-mask churn).
template <int K>
__global__ void wmma_gemm_bias_t(const float* __restrict__ A,
                                 const float* __restrict__ W,
                                 const float* __restrict__ bias,
                                 float* __restrict__ C,
                                 int M, int Ncol) {
  __shared__ float Bp[K * 16];
  const int n0 = blockIdx.x * 16;
  for (int i = threadIdx.x; i < K * 16; i += blockDim.x) {
    int k = i >> 4, j = i & 15;
    Bp[i] = W[k * Ncol + n0 + j];
  }
  __syncthreads();

  const int wave = threadIdx.x >> 5;
  const int lane = threadIdx.x & 31;
  const int half = lane >> 4;     // K-half select for A/B frags
  const int l16  = lane & 15;     // row (A) / col (B)
  const int m0   = blockIdx.y * 64 + wave * 16;
  if (m0 >= M) return;

  v8f acc = {};
  const float* Ar = A + (size_t)(m0 + l16) * K;
#pragma unroll
  for (int k0 = 0; k0 < K; k0 += 4) {
    const int ka = k0 + half * 2;
    v2f a, b;
    if (k0 + 4 <= K) {
      // fast path: whole 4-wide K step in bounds (compile-time for K=128)
      a = *(const v2f*)(Ar + ka);                // global_load_b64
      b.x = Bp[ka * 16 + l16];
      b.y = Bp[(ka + 1) * 16 + l16];
    } else {
      a.x = (ka     < K) ? Ar[ka]                  : 0.0f;
      a.y = (ka + 1 < K) ? Ar[ka + 1]              : 0.0f;
      b.x = (ka     < K) ? Bp[ka * 16 + l16]       : 0.0f;
      b.y = (ka + 1 < K) ? Bp[(ka + 1) * 16 + l16] : 0.0f;
    }
    acc = __builtin_amdgcn_wmma_f32_16x16x4_f32(false, a, false, b,
                                                (short)0, acc, false, false);
  }
  const float bv = bias[n0 + l16];
#pragma unroll
  for (int v = 0; v < 8; ++v) {
    int row = m0 + half * 8 + v;            // lanes 0-15: M=v, lanes 16-31: M=v+8
    C[(size_t)row * Ncol + n0 + l16] = acc[v] + bv;
  }
}

// ---------- edge phase ----------
// wave per edge: score[e,h] = dot(leaky_relu(xl[src,h,:]+xr[tgt,h,:]), att[h,:]); segment max
__global__ void edge_score_kernel(const float* __restrict__ xl,
                                  const float* __restrict__ xr,
                                  const float* __restrict__ att,
                                  const int* __restrict__ ei,
                                  float* __restrict__ score,
                                  unsigned* __restrict__ smax,
                                  int H) {
  int wid  = (blockIdx.x * blockDim.x + threadIdx.x) >> 5;
  int lane = threadIdx.x & 31;
  if (wid >= ETOT) return;
  int s, t; edge_st(ei, wid, s, t);
  const int HC = H * CH;
  for (int h = 0; h < H; ++h) {
    const float* pl = xl + (size_t)s * HC + h * CH;
    const float* pr = xr + (size_t)t * HC + h * CH;
    const float* pa = att + h * CH;
    float acc = 0.0f;
#pragma unroll
    for (int c0 = 0; c0 < CH; c0 += 32) {
      int c = c0 + lane;
      float m = pl[c] + pr[c];
      m = (m > 0.0f) ? m : 0.2f * m;
      acc += m * pa[c];
    }
    for (int off = 16; off > 0; off >>= 1) acc += __shfl_down(acc, off, 32);
    if (lane == 0) {
      score[(size_t)wid * H + h] = acc;
      atomicMax(&smax[t * H + h], fflip(acc));
    }
  }
}

__global__ void edge_exp_kernel(float* __restrict__ score,
                                const unsigned* __restrict__ smax,
                                float* __restrict__ denom,
                                const int* __restrict__ ei, int H) {
  int idx = blockIdx.x * blockDim.x + threadIdx.x;
  if (idx >= ETOT * H) return;
  int e = idx / H, h = idx - e * H;
  int s, t; edge_st(ei, e, s, t); (void)s;
  float a = expf(score[idx] - funflip(smax[t * H + h]));
  score[idx] = a;
  atomicAdd(&denom[t * H + h], a);
}

// wave per edge: out[tgt,c] += (1/H) * sum_h alpha[e,h] * xl[src,h,c]
__global__ void edge_agg_kernel(const float* __restrict__ score,
                                const float* __restrict__ denom,
                                const float* __restrict__ xl,
                                const int* __restrict__ ei,
                                float* __restrict__ out, int H) {
  int wid  = (blockIdx.x * blockDim.x + threadIdx.x) >> 5;
  int lane = threadIdx.x & 31;
  if (wid >= ETOT) return;
  int s, t; edge_st(ei, wid, s, t);
  const int HC = H * CH;
  const float invH = 1.0f / (float)H;
  float* po = out + (size_t)t * CH;
  for (int h = 0; h < H; ++h) {
    float w = invH * score[(size_t)wid * H + h] / denom[t * H + h];
    const float* pl = xl + (size_t)s * HC + h * CH;
#pragma unroll
    for (int c0 = 0; c0 < CH; c0 += 32) {
      int c = c0 + lane;
      atomicAdd(&po[c], w * pl[c]);
    }
  }
}

// ---------- graph norm + gelu ----------
__global__ void bias_stats_kernel(float* __restrict__ h,
                                  const float* __restrict__ bias,
                                  float* __restrict__ stats) {
  int c = threadIdx.x;                       // 128 channels
  float bv = bias[c];
  float s = 0.0f, s2 = 0.0f;
  for (int r = blockIdx.x; r < NNODES; r += gridDim.x) {
    size_t i = (size_t)r * CH + c;
    float v = h[i] + bv;
    h[i] = v;
    s += v; s2 += v * v;
  }
  atomicAdd(&stats[c], s);
  atomicAdd(&stats[CH + c], s2);
}

__global__ void finalize_norm_kernel(float* __restrict__ stats,
                                     const float* __restrict__ gamma,
                                     const float* __restrict__ ms) {
  int c = threadIdx.x;
  float mu  = stats[c] * (1.0f / NNODES);
  float ex2 = stats[CH + c] * (1.0f / NNODES);
  float m   = ms[c];
  float var = ex2 - 2.0f * m * mu * mu + m * m * mu * mu;
  stats[2 * CH + c] = m * mu;                          // shift
  stats[3 * CH + c] = gamma[c] * rsqrtf(var + 1e-5f);  // scale
}

__global__ void norm_gelu_kernel(const float* __restrict__ h,
                                 const float* __restrict__ stats,
                                 const float* __restrict__ beta,
                                 float* __restrict__ out) {
  int idx = blockIdx.x * blockDim.x + threadIdx.x;
  if (idx >= NNODES * CH) return;
  int c = idx & (CH - 1);
  float y = (h[idx] - stats[2 * CH + c]) * stats[3 * CH + c] + beta[c];
  float t = tanhf(0.7978845608028654f * (y + 0.044715f * y * y * y));
  out[idx] = 0.5f * y * (1.0f + t);
}

__global__ void colsum_kernel(const float* __restrict__ h, float* __restrict__ acc) {
  int c = threadIdx.x;
  float s = 0.0f;
  for (int r = blockIdx.x; r < NNODES; r += gridDim.x) s += h[(size_t)r * CH + c];
  atomicAdd(&acc[c], s);
}
__global__ void writeout_kernel(const float* __restrict__ acc, float* __restrict__ out) {
  int c = threadIdx.x;
  out[c] = acc[c] * (1.0f / NNODES);
}

// ---------- host ----------
extern "C" void kernel_launch(void* const* d_in, const int* in_sizes, int n_in,
                              void* d_out, int out_size, void* d_ws, size_t ws_size,
                              hipStream_t stream) {
  (void)in_sizes; (void)n_in; (void)out_size; (void)ws_size;
  const float* x  = (const float*)d_in[0];
  const int*   ei = (const int*)d_in[1];
  const float* p[27];
  for (int i = 0; i < 27; ++i) p[i] = (const float*)d_in[2 + i];
  // p[0..5]=conv1{Wl,bl,Wr,br,att,bias} p[6..8]=norm1{gamma,beta,ms}
  // p[9..14]=conv2, p[15..17]=norm2, p[18..23]=conv3, p[24..26]=norm3

  float* ws      = (float*)d_ws;
  float* xl      = ws;
  float* xr      = xl    + (size_t)NNODES * 256;
  float* hbuf    = xr    + (size_t)NNODES * 256;
  float* hbuf2   = hbuf  + (size_t)NNODES * CH;
  float* score   = hbuf2 + (size_t)NNODES * CH;
  unsigned* smax = (unsigned*)(score + (size_t)ETOT * 2);
  float* denom   = (float*)smax + (size_t)NNODES * 2;
  float* stats   = denom + (size_t)NNODES * 2;   // 512 floats

  auto edges_and_norm = [&](const float* in, int H, const float* const* q) {
    const float* att = q[4]; const float* bias = q[5];
    const float* gamma = q[6]; const float* beta = q[7]; const float* ms = q[8];
    (void)in;
    fill_u32<<<(NNODES * H + 255) / 256, 256, 0, stream>>>(smax, 0u, NNODES * H);
    fill_f32<<<(NNODES * H + 255) / 256, 256, 0, stream>>>(denom, 0.0f, NNODES * H);
    const int wgrid = (ETOT + 7) / 8;            // 8 waves per 256-thread block
    edge_score_kernel<<<wgrid, 256, 0, stream>>>(xl, xr, att, ei, score, smax, H);
    edge_exp_kernel<<<(ETOT * H + 255) / 256, 256, 0, stream>>>(score, smax, denom, ei, H);
    fill_f32<<<(NNODES * CH + 255) / 256, 256, 0, stream>>>(hbuf, 0.0f, NNODES * CH);
    edge_agg_kernel<<<wgrid, 256, 0, stream>>>(score, denom, xl, ei, hbuf, H);
    fill_f32<<<1, 256, 0, stream>>>(stats, 0.0f, 256);
    bias_stats_kernel<<<256, 128, 0, stream>>>(hbuf, bias, stats);
    finalize_norm_kernel<<<1, 128, 0, stream>>>(stats, gamma, ms);
    norm_gelu_kernel<<<(NNODES * CH + 255) / 256, 256, 0, stream>>>(hbuf, stats, beta, hbuf2);
  };

  // ---- layer 1: cin=3, H=2 ----
  {
    dim3 gg(256 / 16, (NNODES + 63) / 64);
    wmma_gemm_bias_t<3><<<gg, 128, 0, stream>>>(x, p[0], p[1], xl, NNODES, 256);
    wmma_gemm_bias_t<3><<<gg, 128, 0, stream>>>(x, p[2], p[3], xr, NNODES, 256);
    edges_and_norm(x, 2, &p[0]);
  }
  // ---- layer 2: cin=128, H=2 ----
  {
    dim3 gg(256 / 16, (NNODES + 63) / 64);
    wmma_gemm_bias_t<128><<<gg, 128, 0, stream>>>(hbuf2, p[9],  p[10], xl, NNODES, 256);
    wmma_gemm_bias_t<128><<<gg, 128, 0, stream>>>(hbuf2, p[11], p[12], xr, NNODES, 256);
    edges_and_norm(hbuf2, 2, &p[9]);
  }
  // ---- layer 3: cin=128, H=1 ----
  {
    dim3 gg(128 / 16, (NNODES + 63) / 64);
    wmma_gemm_bias_t<128><<<gg, 128, 0, stream>>>(hbuf2, p[18], p[19], xl, NNODES, 128);
    wmma_gemm_bias_t<128><<<gg, 128, 0, stream>>>(hbuf2, p[20], p[21], xr, NNODES, 128);
    edges_and_norm(hbuf2, 1, &p[18]);
  }

  fill_f32<<<1, 256, 0, stream>>>(stats, 0.0f, 256);
  colsum_kernel<<<256, 128, 0, stream>>>(hbuf2, stats);
  writeout_kernel<<<1, 128, 0, stream>>>(stats, (float*)d_out);
}